// GraphSAGE_15814069584342
// MI455X (gfx1250) — compile-verified
//
#include <hip/hip_runtime.h>
#include <math.h>

#define NN 100000
#define NE 1600000
#define D  128
#define DO 64

typedef float v2f __attribute__((ext_vector_type(2)));
typedef float v8f __attribute__((ext_vector_type(8)));

// mish(v) = v * tanh(softplus(v)) using hardware v_exp_f32 / v_log_f32.
// sp >= 0 always, so tanh(sp) = (1-e^-2sp)/(1+e^-2sp) is stable.
__device__ inline float mishf(float v) {
    float sp = (v > 20.0f) ? v : __logf(1.0f + __expf(v));
    float e2 = __expf(-2.0f * sp);
    return v * ((1.0f - e2) / (1.0f + e2));
}

__device__ inline void fadd_atomic(float* p, float v) {
    __hip_atomic_fetch_add(p, v, __ATOMIC_RELAXED, __HIP_MEMORY_SCOPE_AGENT);
}

__global__ void zero_kernel(float* __restrict__ p, int n) {
    for (int i = blockIdx.x * blockDim.x + threadIdx.x; i < n; i += gridDim.x * blockDim.x)
        p[i] = 0.0f;
}

// W [rows x cols] row-major -> k-pair-interleaved transpose:
//   WP[(k>>1)*(2*rows) + n*2 + (k&1)] = W[n][k]
// so that B-fragment (k,k+1) for column n is one contiguous float2.
__global__ void transpose_pair_kernel(const float* __restrict__ W, float* __restrict__ WP,
                                      int rows, int cols) {
    int t = blockIdx.x * blockDim.x + threadIdx.x;
    if (t < rows * cols) {
        int n = t / cols, k = t - n * cols;
        WP[(k >> 1) * (2 * rows) + n * 2 + (k & 1)] = W[t];
    }
}

__global__ void count_kernel(const int* __restrict__ dst, float* __restrict__ cnt) {
    int e = blockIdx.x * blockDim.x + threadIdx.x;
    if (e < NE) fadd_atomic(&cnt[dst[e]], 1.0f);
}

// one wave32 per edge; each lane carries float4 (32*4 = 128 features).
// x (51 MB) is L2-resident (192 MB L2), so gather + atomic scatter are L2-bound.
__global__ void __launch_bounds__(256)
scatter_kernel(const float* __restrict__ feat, const int* __restrict__ src,
               const int* __restrict__ dst, float* __restrict__ agg) {
    int wave = blockIdx.x * (blockDim.x >> 5) + (threadIdx.x >> 5);
    int lane = threadIdx.x & 31;
    if (wave >= NE) return;
    int s = src[wave];
    int d = dst[wave];
    const float4 v = *(const float4*)&feat[(long)s * D + lane * 4];
    float* ap = &agg[(long)d * D + lane * 4];
    fadd_atomic(&ap[0], v.x);
    fadd_atomic(&ap[1], v.y);
    fadd_atomic(&ap[2], v.z);
    fadd_atomic(&ap[3], v.w);
}

// Y[m][n] = mish( (A[m]/max(cnt,1)) @ Wl^T + X[m] @ Wr^T + bias )
// block = one 16-row M tile, 8 waves = 8 N-tiles of 16. Full EXEC for WMMA.
__global__ void __launch_bounds__(256)
sage_gemm_kernel(const float* __restrict__ A, const float* __restrict__ X,
                 const float* __restrict__ BPl, const float* __restrict__ BPr,
                 const float* __restrict__ bias, const float* __restrict__ cnt,
                 float* __restrict__ Y) {
    __shared__ float lds_a[16 * 132];
    __shared__ float lds_x[16 * 132];
    const int m0 = blockIdx.x * 16;

    // cooperative tile stage; fuse the mean-divide into A staging
    for (int i = threadIdx.x; i < 16 * 32; i += 256) {
        int row = i >> 5;
        int c4  = (i & 31) * 4;
        float s = 1.0f / fmaxf(cnt[m0 + row], 1.0f);
        float4 va = *(const float4*)&A[(long)(m0 + row) * D + c4];
        va.x *= s; va.y *= s; va.z *= s; va.w *= s;
        *(float4*)&lds_a[row * 132 + c4] = va;
        *(float4*)&lds_x[row * 132 + c4] = *(const float4*)&X[(long)(m0 + row) * D + c4];
    }
    __syncthreads();

    const int wave = threadIdx.x >> 5;
    const int n0   = wave * 16;
    const int lane = threadIdx.x & 31;
    const int col  = lane & 15;
    const int half = lane >> 4;

    v8f c = {};
    // pair-layout B base for this lane's output column
    const float* BL = BPl + (size_t)(n0 + col) * 2;
    const float* BR = BPr + (size_t)(n0 + col) * 2;

#pragma unroll 4
    for (int k0 = 0; k0 < D; k0 += 4) {
        int kk = k0 + 2 * half;          // this half-wave's k-pair start
        int p  = kk >> 1;                // pair index
        v2f a  = *(const v2f*)&lds_a[col * 132 + kk];
        v2f x  = *(const v2f*)&lds_x[col * 132 + kk];
        v2f bl = *(const v2f*)&BL[(size_t)p * (2 * D)];
        v2f br = *(const v2f*)&BR[(size_t)p * (2 * D)];
        c = __builtin_amdgcn_wmma_f32_16x16x4_f32(false, a, false, bl, (short)0, c, false, false);
        c = __builtin_amdgcn_wmma_f32_16x16x4_f32(false, x, false, br, (short)0, c, false, false);
    }

    const float bb = bias[n0 + col];
    const int mbase = m0 + 8 * half;
#pragma unroll
    for (int r = 0; r < 8; ++r)
        Y[(long)(mbase + r) * D + n0 + col] = mishf(c[r] + bb);
}

// out[m][n] = X[m] @ Wh^T + bias  (n < 64); block = 128 threads = 4 waves
__global__ void __launch_bounds__(128)
head_gemm_kernel(const float* __restrict__ X, const float* __restrict__ BP,
                 const float* __restrict__ bias, float* __restrict__ Y) {
    __shared__ float lds_x[16 * 132];
    const int m0 = blockIdx.x * 16;
    for (int i = threadIdx.x; i < 16 * 32; i += 128) {
        int row = i >> 5;
        int c4  = (i & 31) * 4;
        *(float4*)&lds_x[row * 132 + c4] = *(const float4*)&X[(long)(m0 + row) * D + c4];
    }
    __syncthreads();

    const int wave = threadIdx.x >> 5;
    const int n0   = wave * 16;
    const int lane = threadIdx.x & 31;
    const int col  = lane & 15;
    const int half = lane >> 4;

    v8f c = {};
    const float* B = BP + (size_t)(n0 + col) * 2;
#pragma unroll 4
    for (int k0 = 0; k0 < D; k0 += 4) {
        int kk = k0 + 2 * half;
        int p  = kk >> 1;
        v2f x = *(const v2f*)&lds_x[col * 132 + kk];
        v2f b = *(const v2f*)&B[(size_t)p * (2 * DO)];
        c = __builtin_amdgcn_wmma_f32_16x16x4_f32(false, x, false, b, (short)0, c, false, false);
    }
    const float bb = bias[n0 + col];
    const int mbase = m0 + 8 * half;
#pragma unroll
    for (int r = 0; r < 8; ++r)
        Y[(long)(mbase + r) * DO + n0 + col] = c[r] + bb;
}

extern "C" void kernel_launch(void* const* d_in, const int* in_sizes, int n_in,
                              void* d_out, int out_size, void* d_ws, size_t ws_size,
                              hipStream_t stream) {
    const float* x   = (const float*)d_in[0];
    const int*   ei  = (const int*)d_in[1];
    const int*   src = ei;
    const int*   dst = ei + NE;
    const float* Wl0 = (const float*)d_in[2];
    const float* bl0 = (const float*)d_in[3];
    const float* Wr0 = (const float*)d_in[4];
    const float* Wl1 = (const float*)d_in[5];
    const float* bl1 = (const float*)d_in[6];
    const float* Wr1 = (const float*)d_in[7];
    const float* Wh  = (const float*)d_in[8];
    const float* bh  = (const float*)d_in[9];

    float* ws   = (float*)d_ws;
    float* agg  = ws;                            // N*128
    float* h0   = agg  + (size_t)NN * D;         // N*128
    float* h1   = h0   + (size_t)NN * D;         // N*128
    float* cnt  = h1   + (size_t)NN * D;         // N
    float* WP_l0 = cnt   + NN;                   // 128*128 each
    float* WP_r0 = WP_l0 + D * D;
    float* WP_l1 = WP_r0 + D * D;
    float* WP_r1 = WP_l1 + D * D;
    float* WP_h  = WP_r1 + D * D;                // 128*64

    zero_kernel<<<2048, 256, 0, stream>>>(agg, NN * D);
    zero_kernel<<<256, 256, 0, stream>>>(cnt, NN);

    transpose_pair_kernel<<<(D * D + 255) / 256, 256, 0, stream>>>(Wl0, WP_l0, D, D);
    transpose_pair_kernel<<<(D * D + 255) / 256, 256, 0, stream>>>(Wr0, WP_r0, D, D);
    transpose_pair_kernel<<<(D * D + 255) / 256, 256, 0, stream>>>(Wl1, WP_l1, D, D);
    transpose_pair_kernel<<<(D * D + 255) / 256, 256, 0, stream>>>(Wr1, WP_r1, D, D);
    transpose_pair_kernel<<<(DO * D + 255) / 256, 256, 0, stream>>>(Wh, WP_h, DO, D);

    count_kernel<<<(NE + 255) / 256, 256, 0, stream>>>(dst, cnt);

    // layer 0
    scatter_kernel<<<NE / 8, 256, 0, stream>>>(x, src, dst, agg);
    sage_gemm_kernel<<<NN / 16, 256, 0, stream>>>(agg, x, WP_l0, WP_r0, bl0, cnt, h0);

    // layer 1
    zero_kernel<<<2048, 256, 0, stream>>>(agg, NN * D);
    scatter_kernel<<<NE / 8, 256, 0, stream>>>(h0, src, dst, agg);
    sage_gemm_kernel<<<NN / 16, 256, 0, stream>>>(agg, h0, WP_l1, WP_r1, bl1, cnt, h1);

    // head
    head_gemm_kernel<<<NN / 16, 128, 0, stream>>>(h1, WP_h, bh, (float*)d_out);
}